// EHROntologyModel_27805618275297
// MI455X (gfx1250) — compile-verified
//
#include <hip/hip_runtime.h>
#include <math.h>

typedef __attribute__((ext_vector_type(2))) float v2f;
typedef __attribute__((ext_vector_type(4))) float v4f;
typedef __attribute__((ext_vector_type(8))) float v8f;
typedef int v4i_vs __attribute__((vector_size(16)));   // matches builtin param type

// CDNA5 async global->LDS staging path (ASYNCcnt), guarded so compile never breaks.
#if defined(__AMDGCN__) && __has_builtin(__builtin_amdgcn_global_load_async_to_lds_b128) && \
    __has_builtin(__builtin_amdgcn_s_wait_asynccnt)
#define USE_ASYNC_LDS 1
typedef __attribute__((address_space(1))) v4i_vs* gptr_b128;
typedef __attribute__((address_space(3))) v4i_vs* lptr_b128;
#else
#define USE_ASYNC_LDS 0
#endif

#define HID 128
#define NEG_SLOPE 0.2f

// ============================================================================
// GEMM: C[N x 128] = A[N x 128] * W[128 x 128], fp32 via V_WMMA_F32_16X16X4_F32
// grid = N/16 blocks (N guaranteed multiple of 16), block = 256 = 8 wave32s.
// Each wave computes one 16x16 tile of C; A strip staged in LDS (async ASYNCcnt
// copies when available).
// ============================================================================
__global__ void gemm_f32_k128(const float* __restrict__ A,
                              const float* __restrict__ W,
                              float* __restrict__ C, int N)
{
    __shared__ __align__(16) float As[16 * HID];
    const int m0 = blockIdx.x * 16;
    const float* ag = A + (size_t)m0 * HID;   // 16x128 strip is contiguous

    // pull W toward the WGP while we stage A (lowered to global_prefetch_b8)
    __builtin_prefetch(W + ((size_t)threadIdx.x << 6), 0, 1);

#if USE_ASYNC_LDS
    {
        // 2048 floats = 512 x b128 async transfers; 256 threads -> 2 each
        int t0 = threadIdx.x;
        int t1 = threadIdx.x + 256;
        __builtin_amdgcn_global_load_async_to_lds_b128(
            (gptr_b128)(ag + t0 * 4), (lptr_b128)(As + t0 * 4), 0, 0);
        __builtin_amdgcn_global_load_async_to_lds_b128(
            (gptr_b128)(ag + t1 * 4), (lptr_b128)(As + t1 * 4), 0, 0);
        __builtin_amdgcn_s_wait_asynccnt(0);
    }
#else
    for (int t = threadIdx.x; t < 16 * HID; t += 256)
        As[t] = ag[t];
#endif
    __syncthreads();

    const int lane = threadIdx.x & 31;
    const int l16  = lane & 15;
    const int hi   = lane >> 4;                 // 0: K pair {0,1}, 1: K pair {2,3}
    const int c0   = (threadIdx.x >> 5) * 16;   // wave -> 16-column tile
    const int col  = c0 + l16;

    v8f acc = {0.f, 0.f, 0.f, 0.f, 0.f, 0.f, 0.f, 0.f};
#pragma unroll
    for (int k0 = 0; k0 < HID; k0 += 4) {
        const int ka = k0 + (hi << 1);          // even -> 8B aligned ds_load_b64
        // A 16x4 tile: lane(0-15) rows M=0..15 K={0,1}; lane(16-31) K={2,3}
        v2f a = *(const v2f*)(&As[l16 * HID + ka]);
        // B 4x16 tile: row K striped across lanes within a VGPR
        v2f b;
        b.x = W[(size_t)ka * HID + col];
        b.y = W[(size_t)(ka + 1) * HID + col];
        acc = __builtin_amdgcn_wmma_f32_16x16x4_f32(false, a, false, b,
                                                    (short)0, acc, false, false);
    }
    // C/D layout: VGPR r -> M=r (lanes 0-15), M=r+8 (lanes 16-31)
    const int rbase = m0 + (hi << 3);
#pragma unroll
    for (int r = 0; r < 8; ++r)
        C[(size_t)(rbase + r) * HID + col] = acc[r];
}

// ============================================================================
// Elementwise / utility kernels
// ============================================================================
__global__ void fill_f32(float* __restrict__ p, float v, size_t n)
{
    size_t i = (size_t)blockIdx.x * blockDim.x + threadIdx.x;
    if (i < n) p[i] = v;
}

// out[r*128+f] = vec[f]  (one float4 per thread -> b128 traffic)
__global__ void row_broadcast(float* __restrict__ out, const float* __restrict__ vec,
                              size_t rows)
{
    size_t i = (size_t)blockIdx.x * blockDim.x + threadIdx.x;  // i indexes float4s
    if (i >= rows * 32) return;
    ((v4f*)out)[i] = ((const v4f*)vec)[i & 31];
}

// dst[i] = src[map[i]]  (rows of 128 floats, one float4 per thread)
__global__ void gather_rows(const float* __restrict__ src, const int* __restrict__ map,
                            float* __restrict__ dst, size_t rows)
{
    size_t i = (size_t)blockIdx.x * blockDim.x + threadIdx.x;  // i indexes float4s
    if (i >= rows * 32) return;
    size_t r  = i >> 5;
    int    f4 = (int)(i & 31);
    ((v4f*)dst)[i] = ((const v4f*)(src + (size_t)map[r] * HID))[f4];
}

// ============================================================================
// GAT kernels
// ============================================================================
// one wave per node: a_src[i] = h[i].att_src ; a_dst[i] = h[i].att_dst
__global__ void attn_scores(const float* __restrict__ h,
                            const float* __restrict__ av_src,
                            const float* __restrict__ av_dst,
                            float* __restrict__ os, float* __restrict__ od, int N)
{
    int node = (int)(((size_t)blockIdx.x * blockDim.x + threadIdx.x) >> 5);
    int lane = threadIdx.x & 31;
    if (node >= N) return;
    const float* row = h + (size_t)node * HID;
    float s = 0.f, d = 0.f;
    for (int t = lane; t < HID; t += 32) {
        float v = row[t];
        s += v * av_src[t];
        d += v * av_dst[t];
    }
#pragma unroll
    for (int off = 16; off; off >>= 1) {
        s += __shfl_xor(s, off);
        d += __shfl_xor(d, off);
    }
    if (lane == 0) { os[node] = s; od[node] = d; }
}

__device__ __forceinline__ void atomicMaxF(float* addr, float v)
{
    // order-preserving int/uint trick (valid for non-NaN values, -inf init)
    if (v >= 0.f) atomicMax((int*)addr, __float_as_int(v));
    else          atomicMin((unsigned int*)addr, __float_as_uint(v));
}

// pass 1: e = leaky(a_src[src]+a_dst[dst]); seg-max into m[dst]
__global__ void gat_edge_max(const float* __restrict__ as_, const float* __restrict__ ad_,
                             const int* __restrict__ src, const int* __restrict__ dst,
                             int E, int N, float* __restrict__ eb, float* __restrict__ m)
{
    int e = (int)((size_t)blockIdx.x * blockDim.x + threadIdx.x);
    if (e >= E + N) return;
    int s, d;
    if (e < E) { s = src[e]; d = dst[e]; } else { s = d = e - E; }
    float v = as_[s] + ad_[d];
    v = (v > 0.f) ? v : NEG_SLOPE * v;
    eb[e] = v;
    atomicMaxF(&m[d], v);
}

// pass 2: w = exp(e - m[dst]) ; denom[dst] += w
__global__ void gat_edge_exp(float* __restrict__ eb, const float* __restrict__ m,
                             const int* __restrict__ dst, int E, int N,
                             float* __restrict__ dn)
{
    int e = (int)((size_t)blockIdx.x * blockDim.x + threadIdx.x);
    if (e >= E + N) return;
    int d = (e < E) ? dst[e] : (e - E);
    float w = expf(eb[e] - m[d]);
    eb[e] = w;
    atomicAdd(&dn[d], w);
}

// pass 3: out[dst] += h[src] * (w / denom[dst])   (wave per edge)
__global__ void gat_scatter(const float* __restrict__ h, const float* __restrict__ eb,
                            const float* __restrict__ dn,
                            const int* __restrict__ src, const int* __restrict__ dst,
                            int E, int N, float* __restrict__ out)
{
    int e = (int)(((size_t)blockIdx.x * blockDim.x + threadIdx.x) >> 5);
    int lane = threadIdx.x & 31;
    if (e >= E + N) return;
    int s, d;
    if (e < E) { s = src[e]; d = dst[e]; } else { s = d = e - E; }
    float alpha = eb[e] / dn[d];
    const float* hs = h + (size_t)s * HID;
    float*       od = out + (size_t)d * HID;
    for (int t = lane; t < HID; t += 32)
        atomicAdd(&od[t], hs[t] * alpha);
}

// ============================================================================
// GCN kernels
// ============================================================================
__global__ void deg_count(const int* __restrict__ dst, int E, int N,
                          float* __restrict__ deg)
{
    int e = (int)((size_t)blockIdx.x * blockDim.x + threadIdx.x);
    if (e >= E + N) return;
    int d = (e < E) ? dst[e] : (e - E);
    atomicAdd(&deg[d], 1.0f);
}

__global__ void rsqrt_clamp(float* __restrict__ p, int n)
{
    int i = (int)((size_t)blockIdx.x * blockDim.x + threadIdx.x);
    if (i < n) p[i] = rsqrtf(fmaxf(p[i], 1e-12f));
}

// out[dst] += xw[src] * dinv[src]*dinv[dst]  (wave per edge)
__global__ void gcn_scatter(const float* __restrict__ xw, const float* __restrict__ dinv,
                            const int* __restrict__ src, const int* __restrict__ dst,
                            int E, int N, float* __restrict__ out)
{
    int e = (int)(((size_t)blockIdx.x * blockDim.x + threadIdx.x) >> 5);
    int lane = threadIdx.x & 31;
    if (e >= E + N) return;
    int s, d;
    if (e < E) { s = src[e]; d = dst[e]; } else { s = d = e - E; }
    float norm = dinv[s] * dinv[d];
    const float* xs = xw + (size_t)s * HID;
    float*       od = out + (size_t)d * HID;
    for (int t = lane; t < HID; t += 32)
        atomicAdd(&od[t], xs[t] * norm);
}

// ============================================================================
// final-node offsets + cosine similarity
// ============================================================================
__global__ void seg_starts(const int* __restrict__ batch, int n, int* __restrict__ cum)
{
    int i = (int)((size_t)blockIdx.x * blockDim.x + threadIdx.x);
    if (i >= n) return;
    if (i == 0) cum[batch[0]] = 0;
    else if (batch[i] != batch[i - 1]) cum[batch[i]] = i;
}

__global__ void cosine_kernel(const float* __restrict__ lf, const float* __restrict__ rf,
                              const int* __restrict__ cl, const int* __restrict__ cr,
                              float* __restrict__ out, int G)
{
    int g = (int)(((size_t)blockIdx.x * blockDim.x + threadIdx.x) >> 5);
    int lane = threadIdx.x & 31;
    if (g >= G) return;
    const float* a = lf + (size_t)cl[g] * HID;
    const float* b = rf + (size_t)cr[g] * HID;
    float nu = 0.f, na = 0.f, nb = 0.f;
    for (int t = lane; t < HID; t += 32) {
        float x = a[t], y = b[t];
        nu += x * y; na += x * x; nb += y * y;
    }
#pragma unroll
    for (int off = 16; off; off >>= 1) {
        nu += __shfl_xor(nu, off);
        na += __shfl_xor(na, off);
        nb += __shfl_xor(nb, off);
    }
    if (lane == 0) {
        float den = fmaxf(sqrtf(na), 1e-6f) * fmaxf(sqrtf(nb), 1e-6f);
        out[g] = nu / den;
    }
}

// ============================================================================
// host-side orchestration
// ============================================================================
static inline unsigned cdiv(size_t a, size_t b) { return (unsigned)((a + b - 1) / b); }

static void run_gat_layer(const float* x, const float* W, const float* av_src,
                          const float* av_dst, const float* bias,
                          const int* ei, int E, int N,
                          float* h, float* out,
                          float* as_, float* ad_, float* m_, float* dn_, float* eb_,
                          hipStream_t st)
{
    gemm_f32_k128<<<N / 16, 256, 0, st>>>(x, W, h, N);
    attn_scores<<<cdiv((size_t)N * 32, 256), 256, 0, st>>>(h, av_src, av_dst, as_, ad_, N);
    fill_f32<<<cdiv(N, 256), 256, 0, st>>>(m_, -INFINITY, (size_t)N);
    fill_f32<<<cdiv(N, 256), 256, 0, st>>>(dn_, 0.f, (size_t)N);
    int ET = E + N;
    gat_edge_max<<<cdiv(ET, 256), 256, 0, st>>>(as_, ad_, ei, ei + E, E, N, eb_, m_);
    gat_edge_exp<<<cdiv(ET, 256), 256, 0, st>>>(eb_, m_, ei + E, E, N, dn_);
    row_broadcast<<<cdiv((size_t)N * 32, 256), 256, 0, st>>>(out, bias, (size_t)N);
    gat_scatter<<<cdiv((size_t)ET * 32, 256), 256, 0, st>>>(h, eb_, dn_, ei, ei + E, E, N, out);
}

static void run_gcn_layer(const float* x, const float* W, const float* b,
                          const int* ei, int E, int N, const float* dinv,
                          float* xw, float* out, hipStream_t st)
{
    gemm_f32_k128<<<N / 16, 256, 0, st>>>(x, W, xw, N);
    row_broadcast<<<cdiv((size_t)N * 32, 256), 256, 0, st>>>(out, b, (size_t)N);
    int ET = E + N;
    gcn_scatter<<<cdiv((size_t)ET * 32, 256), 256, 0, st>>>(xw, dinv, ei, ei + E, E, N, out);
}

extern "C" void kernel_launch(void* const* d_in, const int* in_sizes, int n_in,
                              void* d_out, int out_size, void* d_ws, size_t ws_size,
                              hipStream_t stream)
{
    (void)n_in; (void)ws_size;
    const float* spec = (const float*)d_in[0];
    const float* tabs[3] = {(const float*)d_in[1], (const float*)d_in[2], (const float*)d_in[3]};

    int Ns[3], E1s[3], E2s[3], Ls[3];
    const float *Ws[3], *asv[3], *adv[3], *bsv[3];
    const int *e1s[3], *e2s[3], *maps[3];
    for (int i = 0; i < 3; ++i) {
        Ns[i]   = in_sizes[1 + i] / HID;
        Ws[i]   = (const float*)d_in[4 + 4 * i];
        asv[i]  = (const float*)d_in[5 + 4 * i];
        adv[i]  = (const float*)d_in[6 + 4 * i];
        bsv[i]  = (const float*)d_in[7 + 4 * i];
        e1s[i]  = (const int*)d_in[22 + 2 * i];
        e2s[i]  = (const int*)d_in[23 + 2 * i];
        E1s[i]  = in_sizes[22 + 2 * i] / 2;
        E2s[i]  = in_sizes[23 + 2 * i] / 2;
        maps[i] = (const int*)d_in[28 + i];
        Ls[i]   = in_sizes[28 + i];
    }
    const float* gW[3] = {(const float*)d_in[16], (const float*)d_in[18], (const float*)d_in[20]};
    const float* gb[3] = {(const float*)d_in[17], (const float*)d_in[19], (const float*)d_in[21]};
    const int* left_x  = (const int*)d_in[31];
    const int* lgi     = (const int*)d_in[32];
    const int* right_x = (const int*)d_in[33];
    const int* rgi     = (const int*)d_in[34];
    const int* lbatch  = (const int*)d_in[35];
    const int* rbatch  = (const int*)d_in[36];
    const int  Np = in_sizes[35];
    const int  Ep = in_sizes[32] / 2;
    const int  G  = out_size;
    float* out = (float*)d_out;

    // ---- workspace carve-up (floats) ----
    size_t maxN = Ns[0]; if ((size_t)Ns[1] > maxN) maxN = Ns[1]; if ((size_t)Ns[2] > maxN) maxN = Ns[2];
    size_t maxET = 0;
    for (int i = 0; i < 3; ++i) {
        size_t e = (size_t)((E1s[i] > E2s[i]) ? E1s[i] : E2s[i]) + Ns[i];
        if (e > maxET) maxET = e;
    }
    size_t ALL = 1 + (size_t)Ls[0] + Ls[1] + Ls[2];

    float* ws = (float*)d_ws;
    size_t off = 0;
    auto alloc = [&](size_t n) { float* p = ws + off; off += n; return p; };
    float* all_emb = alloc(ALL * HID);
    float* b0   = alloc(maxN * HID);   // h scratch
    float* b1   = alloc(maxN * HID);   // layer-1 out
    float* b2   = alloc(maxN * HID);   // layer-2 out
    float* as_  = alloc(maxN);
    float* ad_  = alloc(maxN);
    float* m_   = alloc(maxN);
    float* dn_  = alloc(maxN);
    float* eb_  = alloc(maxET);
    float* lh   = alloc((size_t)Np * HID);
    float* lh2  = alloc((size_t)Np * HID);
    float* rh   = alloc((size_t)Np * HID);
    float* xw   = alloc((size_t)Np * HID);
    float* dinv = alloc((size_t)Np);
    int*   cum  = (int*)alloc(2 * (size_t)G);

    // ---- row 0 of all_emb = spec_emb ----
    (void)hipMemcpyAsync(all_emb, spec, HID * sizeof(float), hipMemcpyDeviceToDevice, stream);

    // ---- ontology GAT stacks ----
    size_t base = 1;
    for (int i = 0; i < 3; ++i) {
        run_gat_layer(tabs[i], Ws[i], asv[i], adv[i], bsv[i], e1s[i], E1s[i], Ns[i],
                      b0, b1, as_, ad_, m_, dn_, eb_, stream);
        run_gat_layer(b1, Ws[i], asv[i], adv[i], bsv[i], e2s[i], E2s[i], Ns[i],
                      b0, b2, as_, ad_, m_, dn_, eb_, stream);
        gather_rows<<<cdiv((size_t)Ls[i] * 32, 256), 256, 0, stream>>>(
            b2, maps[i], all_emb + base * HID, (size_t)Ls[i]);
        base += Ls[i];
    }

    // ---- left patient graph: gather + degrees + 3 GCN layers ----
    gather_rows<<<cdiv((size_t)Np * 32, 256), 256, 0, stream>>>(all_emb, left_x, lh, (size_t)Np);
    fill_f32<<<cdiv(Np, 256), 256, 0, stream>>>(dinv, 0.f, (size_t)Np);
    deg_count<<<cdiv(Ep + Np, 256), 256, 0, stream>>>(lgi + Ep, Ep, Np, dinv);
    rsqrt_clamp<<<cdiv(Np, 256), 256, 0, stream>>>(dinv, Np);
    run_gcn_layer(lh,  gW[0], gb[0], lgi, Ep, Np, dinv, xw, lh2, stream);
    run_gcn_layer(lh2, gW[1], gb[1], lgi, Ep, Np, dinv, xw, lh,  stream);
    run_gcn_layer(lh,  gW[2], gb[2], lgi, Ep, Np, dinv, xw, lh2, stream);  // final left: lh2
    seg_starts<<<cdiv(Np, 256), 256, 0, stream>>>(lbatch, Np, cum);

    // ---- right patient graph (reuse lh as ping buffer) ----
    gather_rows<<<cdiv((size_t)Np * 32, 256), 256, 0, stream>>>(all_emb, right_x, lh, (size_t)Np);
    fill_f32<<<cdiv(Np, 256), 256, 0, stream>>>(dinv, 0.f, (size_t)Np);
    deg_count<<<cdiv(Ep + Np, 256), 256, 0, stream>>>(rgi + Ep, Ep, Np, dinv);
    rsqrt_clamp<<<cdiv(Np, 256), 256, 0, stream>>>(dinv, Np);
    run_gcn_layer(lh, gW[0], gb[0], rgi, Ep, Np, dinv, xw, rh, stream);
    run_gcn_layer(rh, gW[1], gb[1], rgi, Ep, Np, dinv, xw, lh, stream);
    run_gcn_layer(lh, gW[2], gb[2], rgi, Ep, Np, dinv, xw, rh, stream);   // final right: rh
    seg_starts<<<cdiv(Np, 256), 256, 0, stream>>>(rbatch, Np, cum + G);

    // ---- cosine similarity per graph ----
    cosine_kernel<<<cdiv((size_t)G * 32, 256), 256, 0, stream>>>(lh2, rh, cum, cum + G, out, G);
}